// GCNHomogeneous_89584427860362
// MI455X (gfx1250) — compile-verified
//
#include <hip/hip_runtime.h>
#include <hip/hip_bf16.h>

typedef __attribute__((ext_vector_type(2))) float v2f;
typedef __attribute__((ext_vector_type(8))) float v8f;

#define N_NODES   100000
#define N_EDGES   1600000
#define HIDDEN    128
#define NUM_CLASSES 16
#define NUM_GRAPHS  64
#define ROWT      5          // row tiles (of 16) per block: 80 rows/block

// ---------------------------------------------------------------------------
// Degree / normalization
// ---------------------------------------------------------------------------
__global__ void gcn_deg_init(float* __restrict__ deg) {
    int i = blockIdx.x * 256 + threadIdx.x;
    if (i < N_NODES) deg[i] = 1.0f;   // self-loop contributes 1
}

__global__ void gcn_deg_acc(float* __restrict__ deg, const int* __restrict__ dst) {
    int e = blockIdx.x * 256 + threadIdx.x;
    if (e < N_EDGES) atomicAdd(&deg[dst[e]], 1.0f);
}

__global__ void gcn_deg_rsqrt(float* __restrict__ deg) {
    int i = blockIdx.x * 256 + threadIdx.x;
    if (i < N_NODES) deg[i] = rsqrtf(deg[i]);   // deg >= 1 always
}

// ---------------------------------------------------------------------------
// GEMM: Ht = (X @ W) * dinv[row];  Agg initialized to Ht (self-loop term).
// One block = 80 rows x 128 cols; 8 waves, each wave owns one 16-wide column
// tile across 5 row tiles (5 independent WMMA accumulator chains).
// W (128x128, 64KB) staged once per block in LDS; one B fragment feeds 5
// WMMAs. In-place (X aliases Ht for layers >= 1) is safe: __syncthreads()
// separates all X reads from all Ht writes.
// ---------------------------------------------------------------------------
__launch_bounds__(256)
__global__ void gcn_gemm_wmma(const float* X, const float* __restrict__ W,
                              const float* __restrict__ dinv,
                              float* Ht, float* __restrict__ Agg) {
    __shared__ float Wl[HIDDEN * HIDDEN];   // 64 KB of 320 KB WGP LDS

    int tid = threadIdx.x;
    // cooperative 128-bit copy of W into LDS
    const float4* Wv = (const float4*)W;
    float4* Wsv = (float4*)Wl;
    #pragma unroll
    for (int i = 0; i < (HIDDEN * HIDDEN / 4) / 256; ++i)
        Wsv[tid + i * 256] = Wv[tid + i * 256];
    __syncthreads();

    const int row0  = blockIdx.x * (16 * ROWT);
    const int wave  = tid >> 5;          // 0..7 -> 16-wide column tile
    const int lane  = tid & 31;
    const int col0  = wave * 16;
    const int lhalf = lane >> 4;         // 0: K pair {0,1}, 1: K pair {2,3}
    const int l16   = lane & 15;

    v8f acc[ROWT];
    #pragma unroll
    for (int t = 0; t < ROWT; ++t) acc[t] = (v8f){};

    const float* xbase = X + (size_t)(row0 + l16) * HIDDEN + 2 * lhalf;

    #pragma unroll 2
    for (int k = 0; k < HIDDEN; k += 4) {
        // B fragment (4x16 f32): lane half selects K pair, lane%16 = N
        int kk = k + 2 * lhalf;
        v2f b;
        b.x = Wl[kk * HIDDEN + col0 + l16];
        b.y = Wl[(kk + 1) * HIDDEN + col0 + l16];
        #pragma unroll
        for (int t = 0; t < ROWT; ++t) {
            // A fragment (16x4 f32): VGPR0 = K0|K2, VGPR1 = K1|K3 (8B aligned)
            v2f a = *(const v2f*)(xbase + (size_t)t * 16 * HIDDEN + k);
            acc[t] = __builtin_amdgcn_wmma_f32_16x16x4_f32(
                         false, a, false, b, (short)0, acc[t], false, false);
        }
    }

    __syncthreads();   // all X reads complete before any in-place Ht write

    // C/D layout: VGPR r = row (r + 8*lhalf), lane%16 = col
    #pragma unroll
    for (int t = 0; t < ROWT; ++t) {
        #pragma unroll
        for (int r = 0; r < 8; ++r) {
            int row = row0 + t * 16 + r + 8 * lhalf;
            float v = acc[t][r] * dinv[row];
            size_t o = (size_t)row * HIDDEN + col0 + l16;
            Ht[o]  = v;
            Agg[o] = v;   // self-loop: agg starts at ht
        }
    }
}

// ---------------------------------------------------------------------------
// Edge scatter: one wave per edge; lane owns 4 channels (float4 gather,
// 4x global_atomic_add_f32 into the L2-resident accumulator).
// ---------------------------------------------------------------------------
__launch_bounds__(256)
__global__ void gcn_edge_scatter(const float4* __restrict__ Ht4,
                                 float* __restrict__ Agg,
                                 const int* __restrict__ src,
                                 const int* __restrict__ dst) {
    int gw   = (blockIdx.x * 256 + threadIdx.x) >> 5;
    int lane = threadIdx.x & 31;
    int nw   = (gridDim.x * 256) >> 5;
    for (int e = gw; e < N_EDGES; e += nw) {
        int s = src[e];
        int d = dst[e];
        float4 v = Ht4[(size_t)s * 32 + lane];          // 128 ch = 32 float4
        float* out = Agg + (size_t)d * HIDDEN + lane * 4;
        atomicAdd(out + 0, v.x);
        atomicAdd(out + 1, v.y);
        atomicAdd(out + 2, v.z);
        atomicAdd(out + 3, v.w);
    }
}

// ---------------------------------------------------------------------------
// Epilogue: Xn = [relu]( dinv[node] * Agg + bias[ch] )
// ---------------------------------------------------------------------------
__global__ void gcn_epilogue(const float* __restrict__ Agg,
                             const float* __restrict__ dinv,
                             const float* __restrict__ bias,
                             float* __restrict__ Xn, int relu) {
    int i = blockIdx.x * 256 + threadIdx.x;            // N_NODES*HIDDEN exact
    float v = dinv[i >> 7] * Agg[i] + bias[i & 127];
    if (relu) v = fmaxf(v, 0.0f);
    Xn[i] = v;
}

// ---------------------------------------------------------------------------
// Mean pool + classifier head
// ---------------------------------------------------------------------------
__global__ void gcn_pool_zero(float* __restrict__ sums, float* __restrict__ cnt) {
    int i = blockIdx.x * 256 + threadIdx.x;
    if (i < NUM_GRAPHS * HIDDEN) sums[i] = 0.0f;
    if (i < NUM_GRAPHS) cnt[i] = 0.0f;
}

__launch_bounds__(256)
__global__ void gcn_pool_acc(const float4* __restrict__ X4,
                             const int* __restrict__ batch,
                             float* __restrict__ sums, float* __restrict__ cnt) {
    int gw   = (blockIdx.x * 256 + threadIdx.x) >> 5;
    int lane = threadIdx.x & 31;
    int nw   = (gridDim.x * 256) >> 5;
    for (int n = gw; n < N_NODES; n += nw) {
        int g = batch[n];
        float4 v = X4[(size_t)n * 32 + lane];
        float* s = sums + (size_t)g * HIDDEN + lane * 4;
        atomicAdd(s + 0, v.x);
        atomicAdd(s + 1, v.y);
        atomicAdd(s + 2, v.z);
        atomicAdd(s + 3, v.w);
        if (lane == 0) atomicAdd(&cnt[g], 1.0f);
    }
}

__global__ void gcn_head(const float* __restrict__ sums,
                         const float* __restrict__ cnt,
                         const float* __restrict__ linW,
                         const float* __restrict__ linb,
                         float* __restrict__ out) {
    int i = blockIdx.x * 256 + threadIdx.x;
    if (i >= NUM_GRAPHS * NUM_CLASSES) return;
    int g = i >> 4, c = i & 15;
    float inv = 1.0f / fmaxf(cnt[g], 1.0f);
    float acc = 0.0f;
    #pragma unroll 8
    for (int k = 0; k < HIDDEN; ++k)
        acc += sums[g * HIDDEN + k] * linW[k * NUM_CLASSES + c];
    out[i] = acc * inv + linb[c];
}

// ---------------------------------------------------------------------------
// Host launch
// ---------------------------------------------------------------------------
extern "C" void kernel_launch(void* const* d_in, const int* in_sizes, int n_in,
                              void* d_out, int out_size, void* d_ws, size_t ws_size,
                              hipStream_t stream) {
    const float* x     = (const float*)d_in[0];           // [N,128]
    const int*   ei    = (const int*)d_in[1];             // [2,E]
    const int*   batch = (const int*)d_in[2];             // [N]
    const float* Ws    = (const float*)d_in[3];           // [3,128,128]
    const float* bs    = (const float*)d_in[4];           // [3,128]
    const float* linW  = (const float*)d_in[5];           // [128,16]
    const float* linb  = (const float*)d_in[6];           // [16]
    float* out = (float*)d_out;                           // [64,16]

    const int* src = ei;
    const int* dst = ei + N_EDGES;

    // workspace layout (floats, 128-element aligned)
    float* ws   = (float*)d_ws;
    float* dinv = ws;                                     // 100000 (pad 100352)
    float* A    = ws + 100352;                            // ht / x ping buffer
    float* B    = A + (size_t)N_NODES * HIDDEN;           // agg
    float* sums = B + (size_t)N_NODES * HIDDEN;           // 64*128
    float* cnt  = sums + NUM_GRAPHS * HIDDEN;             // 64

    const int nodeBlocks = (N_NODES + 255) / 256;         // 391
    const int gemmBlocks = N_NODES / (16 * ROWT);         // 1250 (exact)
    const int edgeThr    = N_EDGES / 256;                 // 6250 (exact)
    const int ewBlocks   = 4096;                          // edge-scatter waves
    const int elemBlocks = N_NODES * HIDDEN / 256;        // 50000 (exact)

    // D^{-1/2}
    gcn_deg_init<<<nodeBlocks, 256, 0, stream>>>(dinv);
    gcn_deg_acc<<<edgeThr, 256, 0, stream>>>(dinv, dst);
    gcn_deg_rsqrt<<<nodeBlocks, 256, 0, stream>>>(dinv);

    // 3 GCN layers (ReLU on first two)
    for (int l = 0; l < 3; ++l) {
        const float* xin = (l == 0) ? x : A;
        gcn_gemm_wmma<<<gemmBlocks, 256, 0, stream>>>(
            xin, Ws + (size_t)l * HIDDEN * HIDDEN, dinv, A, B);
        gcn_edge_scatter<<<ewBlocks, 256, 0, stream>>>(
            (const float4*)A, B, src, dst);
        gcn_epilogue<<<elemBlocks, 256, 0, stream>>>(
            B, dinv, bs + (size_t)l * HIDDEN, A, (l < 2) ? 1 : 0);
    }

    // global mean pool + linear head
    gcn_pool_zero<<<(NUM_GRAPHS * HIDDEN + 255) / 256, 256, 0, stream>>>(sums, cnt);
    gcn_pool_acc<<<1024, 256, 0, stream>>>((const float4*)A, batch, sums, cnt);
    gcn_head<<<(NUM_GRAPHS * NUM_CLASSES + 255) / 256, 256, 0, stream>>>(
        sums, cnt, linW, linb, out);
}